// MyLayer_76862734729569
// MI455X (gfx1250) — compile-verified
//
#include <hip/hip_runtime.h>

#define BATCH    65536
#define IN_DIM   256
#define FDIM     32
#define ODIM     30
#define NTILES   16   // 256 / 16 columns of M
#define KTILES   64   // 256 / 4 K-steps

typedef __attribute__((ext_vector_type(2))) float v2f;
typedef __attribute__((ext_vector_type(8))) float v8f;

// ---------------------------------------------------------------------------
// Kernel 1: build M = triu((kernel @ kernel^T) * fw, k=1) directly in WMMA
// B-fragment order.  ws element index:
//     idx = ((ntile*KTILES + kt)*32 + lane)*2 + r
// holding M[4*kt + 2*(lane/16) + r][ntile*16 + (lane%16)]
// (B 4x16 layout: k = 2*(lane/16) + reg, n = lane%16)
// ---------------------------------------------------------------------------
__global__ void build_M_frag(const float* __restrict__ kern,   // [256][30]
                             const float* __restrict__ fwm,    // [32][32]
                             const int*   __restrict__ fids,   // [256]
                             float*       __restrict__ ws) {
    int t = blockIdx.x * blockDim.x + threadIdx.x;   // 0..65535
    int r     =  t        & 1;
    int lane  = (t >> 1)  & 31;
    int kt    = (t >> 6)  & 63;
    int ntile =  t >> 12;

    int i = 4 * kt + 2 * (lane >> 4) + r;   // row of M (K dim of GEMM)
    int j = ntile * 16 + (lane & 15);       // col of M (N dim of GEMM)

    float v = 0.0f;
    if (j > i) {                            // strict upper triangle
        float acc = 0.0f;
        const float* ki = kern + i * ODIM;
        const float* kj = kern + j * ODIM;
        #pragma unroll
        for (int c = 0; c < ODIM; ++c) acc += ki[c] * kj[c];
        v = acc * fwm[fids[i] * FDIM + fids[j]];
    }
    ws[t] = v;
}

// ---------------------------------------------------------------------------
// Kernel 2: out[b] = x_b^T M x_b using V_WMMA_F32_16X16X4_F32.
// One wave handles 16 batch rows; A fragments (X tile) pinned in 128 VGPRs.
// ---------------------------------------------------------------------------
__global__ void __launch_bounds__(256)
quadform_wmma(const float* __restrict__ X,      // [65536][256]
              const float* __restrict__ Mfrag,  // fragment-ordered M
              float*       __restrict__ out) {  // [65536]
    const int lane = threadIdx.x & 31;
    const int wave = threadIdx.x >> 5;
    const int h    = lane >> 4;      // half-wave: selects K pair / row group
    const int ln   = lane & 15;
    const int base = (blockIdx.x * 8 + wave) * 16;   // batch tile base row

    // ---- Load all 64 A fragments once: a[kt] = X[base+ln][4*kt + 2h + {0,1}]
    v2f a[KTILES];
    const float* xrow = X + (size_t)(base + ln) * IN_DIM + 2 * h;
    #pragma unroll
    for (int kt = 0; kt < KTILES; ++kt) {
        a[kt] = *(const v2f*)(xrow + 4 * kt);        // global_load_b64
    }

    float racc[8];
    #pragma unroll
    for (int v = 0; v < 8; ++v) racc[v] = 0.0f;

    const v2f* bbase = (const v2f*)Mfrag;

    for (int nt = 0; nt < NTILES; ++nt) {
        v8f c = {0.f, 0.f, 0.f, 0.f, 0.f, 0.f, 0.f, 0.f};

        // C_tile += X_tile(16x256) @ M_panel(256x16), 4 K per WMMA
        #pragma unroll
        for (int kt = 0; kt < KTILES; ++kt) {
            v2f b = bbase[(size_t)(nt * KTILES + kt) * 32 + lane];
            c = __builtin_amdgcn_wmma_f32_16x16x4_f32(
                    /*neg_a=*/false, a[kt],
                    /*neg_b=*/false, b,
                    /*c_mod=*/(short)0, c,
                    /*reuse_a=*/false, /*reuse_b=*/false);
        }

        // Epilogue: racc[v] += C[row][col] * X[row][col]
        // lane holds C[base + v + 8h][nt*16 + ln]  -> coalesced 64B X reloads
        #pragma unroll
        for (int v = 0; v < 8; ++v) {
            float xv = X[(size_t)(base + v + 8 * h) * IN_DIM + nt * 16 + ln];
            racc[v] += c[v] * xv;
        }
    }

    // ---- Reduce over the 16 lanes of each half (masks < 16 stay in-half)
    #pragma unroll
    for (int v = 0; v < 8; ++v) {
        float s = racc[v];
        s += __shfl_xor(s, 1, 32);
        s += __shfl_xor(s, 2, 32);
        s += __shfl_xor(s, 4, 32);
        s += __shfl_xor(s, 8, 32);
        if (ln == 0) out[base + v + 8 * h] = s;   // lanes 0 and 16 write
    }
}

// ---------------------------------------------------------------------------
extern "C" void kernel_launch(void* const* d_in, const int* in_sizes, int n_in,
                              void* d_out, int out_size, void* d_ws, size_t ws_size,
                              hipStream_t stream) {
    const float* x    = (const float*)d_in[0];   // (65536, 256) f32
    const float* kern = (const float*)d_in[1];   // (256, 30)    f32
    const float* fwm  = (const float*)d_in[2];   // (32, 32)     f32
    const int*   fids = (const int*)  d_in[3];   // (256,)       i32
    float*       ws   = (float*)d_ws;            // needs 256 KB
    float*       o    = (float*)d_out;           // (65536,)     f32

    (void)in_sizes; (void)n_in; (void)out_size; (void)ws_size;

    // Build fragment-ordered M (65536 elements, one thread each)
    build_M_frag<<<256, 256, 0, stream>>>(kern, fwm, fids, ws);

    // 4096 waves * 16 rows = 65536 batch rows; 8 waves per block
    quadform_wmma<<<512, 256, 0, stream>>>(x, ws, o);
}